// BasePllay_67207648248502
// MI455X (gfx1250) — compile-verified
//
#include <hip/hip_runtime.h>
#include <math.h>

// ---- problem constants (match reference) ----
#define GRIDN 28
#define NPTS  784        // GRIDN*GRIDN
#define BATCH 128
#define TT    25
#define KMAX  2
#define FOUT  50         // T*K_MAX == F_OUT == 50
#define KPAD  52         // FOUT padded to multiple of 4 for 16x16x4 steps
#define NCLS  10
#define M0C   0.05f

typedef __attribute__((ext_vector_type(2))) float v2f;
typedef __attribute__((ext_vector_type(8))) float v8f;

#if defined(__gfx1250__) && __has_builtin(__builtin_amdgcn_wmma_f32_16x16x4_f32)
#define HAVE_WMMA_F32X4 1
#else
#define HAVE_WMMA_F32X4 0
#endif

// ---------------------------------------------------------------------------
// K1: regenerate grid constants. pts[p] = (27 - p/28, p%28) so
// D2[p][q] = (ip-iq)^2 + (jp-jq)^2 exactly (integer). Stable counting-rank
// sort per row; tie order is irrelevant to the water-filling sum.
// ---------------------------------------------------------------------------
__global__ void grid_consts_kernel(float* __restrict__ sorted_d2,
                                   int* __restrict__ sort_idx) {
    __shared__ int d2row[NPTS];
    const int p  = blockIdx.x;
    const int ip = p / GRIDN, jp = p % GRIDN;
    for (int q = threadIdx.x; q < NPTS; q += 256) {
        int iq = q / GRIDN, jq = q % GRIDN;
        int di = ip - iq, dj = jp - jq;
        d2row[q] = di * di + dj * dj;
    }
    __syncthreads();
    for (int q = threadIdx.x; q < NPTS; q += 256) {
        int my = d2row[q];
        int rank = 0;
        for (int r = 0; r < NPTS; ++r) {
            int v = d2row[r];
            rank += (v < my) | ((v == my) & (r < q));
        }
        sorted_d2[(size_t)p * NPTS + rank] = (float)my;
        sort_idx [(size_t)p * NPTS + rank] = q;
    }
}

// ---------------------------------------------------------------------------
// K2: distance-to-measure. One block per batch; weights in LDS;
// bound = 0.05 * sum(w); water-filling scan with deterministic early exit.
// ---------------------------------------------------------------------------
__global__ void dtm_kernel(const float* __restrict__ w,
                           const float* __restrict__ sorted_d2,
                           const int* __restrict__ sort_idx,
                           float* __restrict__ dtm) {
    __shared__ float wsh[NPTS];
    __shared__ float red[256];
    const int b = blockIdx.x;
    float local = 0.f;
    for (int p = threadIdx.x; p < NPTS; p += 256) {
        float v = w[(size_t)b * NPTS + p];
        wsh[p] = v;
        local += v;
    }
    red[threadIdx.x] = local;
    __syncthreads();
    for (int off = 128; off > 0; off >>= 1) {
        if (threadIdx.x < off) red[threadIdx.x] += red[threadIdx.x + off];
        __syncthreads();
    }
    const float bound = M0C * red[0];
    for (int p = threadIdx.x; p < NPTS; p += 256) {
        const float* sd2  = sorted_d2 + (size_t)p * NPTS;
        const int*   sidx = sort_idx  + (size_t)p * NPTS;
        __builtin_prefetch(sidx, 0, 0);   // global_prefetch_b8
        float cum = 0.f, acc = 0.f;
        for (int k = 0; k < NPTS; ++k) {
            float wj   = wsh[sidx[k]];
            float prev = cum;
            cum += wj;
            float term = (cum < bound) ? wj
                       : ((prev < bound) ? (bound - prev) : 0.f);
            acc += term * sd2[k];
            if (cum >= bound) break;      // all subsequent terms are 0
        }
        float dtm2 = acc / bound;
        dtm[(size_t)b * NPTS + p] = sqrtf(fmaxf(dtm2, 0.f));
    }
}

// ---------------------------------------------------------------------------
// K3: persistence landscape. One block per batch. 4-neighbor edge-clamped
// max, then per-threshold top-2 via per-thread top2 + LDS pair-merge.
// ---------------------------------------------------------------------------
__global__ void landscape_kernel(const float* __restrict__ dtm,
                                 float* __restrict__ land) {
    __shared__ float f[NPTS];
    __shared__ float dmax[NPTS];
    __shared__ float s1[256], s2[256];
    const int b = blockIdx.x;
    for (int p = threadIdx.x; p < NPTS; p += 256)
        f[p] = dtm[(size_t)b * NPTS + p];
    __syncthreads();
    for (int p = threadIdx.x; p < NPTS; p += 256) {
        int i = p / GRIDN, j = p % GRIDN;
        float g  = f[p];
        float up = f[max(i - 1, 0) * GRIDN + j];
        float dn = f[min(i + 1, GRIDN - 1) * GRIDN + j];
        float lf = f[i * GRIDN + max(j - 1, 0)];
        float rt = f[i * GRIDN + min(j + 1, GRIDN - 1)];
        dmax[p] = fmaxf(fmaxf(fmaxf(up, dn), fmaxf(lf, rt)), g);
    }
    __syncthreads();
    for (int t = 0; t < TT; ++t) {
        float tv = 2.0f * (float)t / (float)(TT - 1);
        float v1 = -1e30f, v2 = -1e30f;
        for (int p = threadIdx.x; p < NPTS; p += 256) {
            float tent = fmaxf(fminf(tv - f[p], dmax[p] - tv), 0.f);
            if (tent > v1) { v2 = v1; v1 = tent; }
            else if (tent > v2) { v2 = tent; }
        }
        s1[threadIdx.x] = v1;
        s2[threadIdx.x] = v2;
        __syncthreads();
        for (int off = 128; off > 0; off >>= 1) {
            if (threadIdx.x < off) {
                float a1 = s1[threadIdx.x],       a2 = s2[threadIdx.x];
                float b1 = s1[threadIdx.x + off], b2 = s2[threadIdx.x + off];
                s1[threadIdx.x] = fmaxf(a1, b1);
                s2[threadIdx.x] = fmaxf(fminf(a1, b1), fmaxf(a2, b2));
            }
            __syncthreads();
        }
        if (threadIdx.x == 0) {
            land[(size_t)b * FOUT + t * 2 + 0] = s1[0];
            land[(size_t)b * FOUT + t * 2 + 1] = s2[0];
        }
        __syncthreads();
    }
}

// ---------------------------------------------------------------------------
// Shared WMMA tile engine: As[16][KPAD] row-major (zero-padded K), and B
// staged TRANSPOSED as Bt[16][KPAD] so both fragment reads are contiguous
// 8-byte LDS loads (ds_load_b64), branch-free, EXEC all-1s across the wmma.
// ISA 16x4 layout: lanes0-15 -> {K=k0,k0+1}, lanes16-31 -> {K=k0+2,k0+3};
// C/D: VGPR r -> rows r / r+8 split across lane halves.
// ---------------------------------------------------------------------------
__device__ __forceinline__ v8f wmma_tile_f32(const float (*As)[KPAD],
                                             const float (*Bt)[KPAD],
                                             int lane) {
    const int half = lane >> 4;
    const int l16  = lane & 15;
    v8f c = {};
#if HAVE_WMMA_F32X4
#pragma unroll
    for (int k0 = 0; k0 < KPAD; k0 += 4) {
        int ka = k0 + half * 2;
        v2f a  = *(const v2f*)&As[l16][ka];
        v2f bb = *(const v2f*)&Bt[l16][ka];
        c = __builtin_amdgcn_wmma_f32_16x16x4_f32(false, a, false, bb,
                                                  (short)0, c, false, false);
    }
#else
    for (int r = 0; r < 8; ++r) {
        float acc = 0.f;
        int m = half * 8 + r;
        for (int k = 0; k < KPAD; ++k) acc += As[m][k] * Bt[l16][k];
        c[r] = acc;
    }
#endif
    return c;
}

// ---------------------------------------------------------------------------
// K4: x1 = land(128x50) @ W_topo(50x50) + b_topo, FP32 WMMA.
// One wave per 16x16 tile; grid (8,4). Staging uses clamped-address loads +
// select (v_cndmask) so no exec-mask branches anywhere.
// ---------------------------------------------------------------------------
__global__ void gemm_topo_kernel(const float* __restrict__ A,    // 128x50
                                 const float* __restrict__ Wt,   // 50x50
                                 const float* __restrict__ bias, // 50
                                 float* __restrict__ X1) {       // 128x50
    __shared__ float As[16][KPAD];
    __shared__ float Bt[16][KPAD];
    const int tm = blockIdx.x;      // 0..7
    const int tn = blockIdx.y;      // 0..3
    const int lane = threadIdx.x;   // wave32

    // stage A tile (rows m = tm*16 + r, cols k; k>=50 -> 0)
    for (int idx = lane; idx < 16 * KPAD; idx += 32) {
        int r = idx / KPAD, k = idx % KPAD;
        int m  = tm * 16 + r;
        int kk = min(k, FOUT - 1);
        float v = A[(size_t)m * FOUT + kk];         // always in-bounds
        As[r][k] = (k < FOUT) ? v : 0.f;
    }
    // stage B tile transposed: Bt[n][k] = W_topo[k][tn*16+n] (OOB -> 0)
    for (int idx = lane; idx < 16 * KPAD; idx += 32) {
        int n = idx / KPAD, k = idx % KPAD;
        int gn = min(tn * 16 + n, FOUT - 1);
        int kk = min(k, FOUT - 1);
        float v = Wt[(size_t)kk * FOUT + gn];       // always in-bounds
        Bt[n][k] = ((k < FOUT) && (tn * 16 + n < FOUT)) ? v : 0.f;
    }
    __syncthreads();

    v8f c = wmma_tile_f32(As, Bt, lane);

    const int half = lane >> 4;
    const int l16  = lane & 15;
    int n = tn * 16 + l16;
    if (n < FOUT) {
        float bn = bias[n];
        for (int r = 0; r < 8; ++r) {
            int m = tm * 16 + half * 8 + r;
            X1[(size_t)m * FOUT + n] = c[r] + bn;
        }
    }
}

// ---------------------------------------------------------------------------
// K5: per-feature stats: signal = sum_b |x1|, mu, rstd (ddof=0 variance).
// ---------------------------------------------------------------------------
__global__ void stats_kernel(const float* __restrict__ X1,
                             float* __restrict__ signal_out,
                             float* __restrict__ mu,
                             float* __restrict__ rstd) {
    int f = threadIdx.x;
    if (f >= FOUT) return;
    float s = 0.f, s2 = 0.f, sa = 0.f;
    for (int b = 0; b < BATCH; ++b) {
        float v = X1[(size_t)b * FOUT + f];
        s += v; s2 += v * v; sa += fabsf(v);
    }
    float m   = s  / (float)BATCH;
    float var = s2 / (float)BATCH - m * m;
    mu[f]   = m;
    rstd[f] = rsqrtf(var + 1e-5f);
    signal_out[f] = sa;
}

// ---------------------------------------------------------------------------
// K6: out = relu(BN(x1)) @ W_fc(50x10) + b_fc, WMMA. BN+ReLU applied ONCE
// during LDS staging of the A tile (not per fragment read).
// ---------------------------------------------------------------------------
__global__ void gemm_fc_kernel(const float* __restrict__ X1,
                               const float* __restrict__ gamma,
                               const float* __restrict__ beta,
                               const float* __restrict__ mu,
                               const float* __restrict__ rstd,
                               const float* __restrict__ Wfc,  // 50x10
                               const float* __restrict__ bfc,  // 10
                               float* __restrict__ out) {      // 128x10
    __shared__ float As[16][KPAD];
    __shared__ float Bt[16][KPAD];
    const int tm = blockIdx.x;      // 0..7
    const int lane = threadIdx.x;

    // stage BN+ReLU A tile
    for (int idx = lane; idx < 16 * KPAD; idx += 32) {
        int r = idx / KPAD, k = idx % KPAD;
        int m  = tm * 16 + r;
        int kk = min(k, FOUT - 1);
        float x = X1[(size_t)m * FOUT + kk];
        float y = gamma[kk] * (x - mu[kk]) * rstd[kk] + beta[kk];
        y = fmaxf(y, 0.f);
        As[r][k] = (k < FOUT) ? y : 0.f;
    }
    // stage W_fc transposed: Bt[n][k] = W_fc[k][n] (n>=10 or k>=50 -> 0)
    for (int idx = lane; idx < 16 * KPAD; idx += 32) {
        int n = idx / KPAD, k = idx % KPAD;
        int gn = min(n, NCLS - 1);
        int kk = min(k, FOUT - 1);
        float v = Wfc[(size_t)kk * NCLS + gn];
        Bt[n][k] = ((k < FOUT) && (n < NCLS)) ? v : 0.f;
    }
    __syncthreads();

    v8f c = wmma_tile_f32(As, Bt, lane);

    const int half = lane >> 4;
    const int l16  = lane & 15;
    int n = l16;
    if (n < NCLS) {
        float bn = bfc[n];
        for (int r = 0; r < 8; ++r) {
            int m = tm * 16 + half * 8 + r;
            out[(size_t)m * NCLS + n] = c[r] + bn;
        }
    }
}

// ---------------------------------------------------------------------------
// Launch: deterministic sequence, all on `stream`, scratch carved from d_ws.
// ws usage: 784*784*(4+4) + 128*784*4 + small ~= 5.4 MB.
// ---------------------------------------------------------------------------
extern "C" void kernel_launch(void* const* d_in, const int* in_sizes, int n_in,
                              void* d_out, int out_size, void* d_ws, size_t ws_size,
                              hipStream_t stream) {
    const float* input  = (const float*)d_in[0];   // 128*784
    const float* W_topo = (const float*)d_in[1];   // 50*50
    const float* b_topo = (const float*)d_in[2];   // 50
    const float* gamma  = (const float*)d_in[3];   // 50
    const float* beta   = (const float*)d_in[4];   // 50
    const float* W_fc   = (const float*)d_in[5];   // 50*10
    const float* b_fc   = (const float*)d_in[6];   // 10

    float* out    = (float*)d_out;                 // 128*10
    float* signal = out + BATCH * NCLS;            // +50

    char* ws = (char*)d_ws;
    float* sorted_d2 = (float*)ws;  ws += sizeof(float) * (size_t)NPTS * NPTS;
    int*   sort_idx  = (int*)ws;    ws += sizeof(int)   * (size_t)NPTS * NPTS;
    float* dtm       = (float*)ws;  ws += sizeof(float) * (size_t)BATCH * NPTS;
    float* land      = (float*)ws;  ws += sizeof(float) * (size_t)BATCH * FOUT;
    float* x1        = (float*)ws;  ws += sizeof(float) * (size_t)BATCH * FOUT;
    float* mu        = (float*)ws;  ws += sizeof(float) * 64;
    float* rstd      = (float*)ws;  ws += sizeof(float) * 64;
    (void)ws; (void)ws_size; (void)in_sizes; (void)n_in; (void)out_size;

    grid_consts_kernel<<<NPTS, 256, 0, stream>>>(sorted_d2, sort_idx);
    dtm_kernel<<<BATCH, 256, 0, stream>>>(input, sorted_d2, sort_idx, dtm);
    landscape_kernel<<<BATCH, 256, 0, stream>>>(dtm, land);
    gemm_topo_kernel<<<dim3(8, 4), 32, 0, stream>>>(land, W_topo, b_topo, x1);
    stats_kernel<<<1, 64, 0, stream>>>(x1, signal, mu, rstd);
    gemm_fc_kernel<<<8, 32, 0, stream>>>(x1, gamma, beta, mu, rstd, W_fc, b_fc, out);
}